// NodeInfoScoreLayer_58394375356876
// MI455X (gfx1250) — compile-verified
//
#include <hip/hip_runtime.h>
#include <hip/hip_bf16.h>

typedef __attribute__((ext_vector_type(2))) float v2f;
typedef __attribute__((ext_vector_type(8))) float v8f;

#define D_FEAT 96

// ---------------------------------------------------------------------------
// 0) zero workspace (degrees + agg)
// ---------------------------------------------------------------------------
__global__ void zero_kernel(float* __restrict__ p, long long n) {
    long long i = (long long)blockIdx.x * blockDim.x + threadIdx.x;
    if (i < n) p[i] = 0.0f;
}

// ---------------------------------------------------------------------------
// 1) degree counts: out_deg[src[e]] += 1, in_deg[dst[e]] += 1
//    counts < 2^24 so float accumulation is exact
// ---------------------------------------------------------------------------
__global__ void degree_kernel(const int* __restrict__ src,
                              const int* __restrict__ dst,
                              float* __restrict__ out_deg,
                              float* __restrict__ in_deg,
                              int n_edges) {
    int i = blockIdx.x * blockDim.x + threadIdx.x;
    int stride = gridDim.x * blockDim.x;
    for (int e = i; e < n_edges; e += stride) {
        atomicAdd(&out_deg[src[e]], 1.0f);
        atomicAdd(&in_deg[dst[e]], 1.0f);
    }
}

// ---------------------------------------------------------------------------
// 2) normalized scatter-add SpMM: one wave32 per edge.
//    Lane k handles feature elements k, k+32, k+64 -> three fully-coalesced
//    128B segments per row. Non-returning global_atomic_add_f32 into agg,
//    which is L2-resident (19.2 MB << 192 MB L2).
// ---------------------------------------------------------------------------
__global__ void scatter_kernel(const float* __restrict__ feat,
                               const int* __restrict__ src,
                               const int* __restrict__ dst,
                               const float* __restrict__ out_deg,
                               float* __restrict__ agg,
                               int n_edges) {
    int lane   = threadIdx.x & 31;
    int wave   = (blockIdx.x * blockDim.x + threadIdx.x) >> 5;
    int nwaves = (gridDim.x * blockDim.x) >> 5;

    for (int e = wave; e < n_edges; e += nwaves) {
        int s = src[e];
        int d = dst[e];

        // prefetch next edge's source row to hide random-gather latency
        int en = e + nwaves;
        if (en < n_edges) {
            int sn = src[en];
            __builtin_prefetch(&feat[(long long)sn * D_FEAT], 0, 1);
        }

        float norm = rsqrtf(fmaxf(out_deg[s], 1.0f));
        const float* __restrict__ frow = feat + (long long)s * D_FEAT;
        float* __restrict__ arow       = agg  + (long long)d * D_FEAT;

#pragma unroll
        for (int j = 0; j < 3; ++j) {
            float v = frow[lane + 32 * j] * norm;
            atomicAdd(&arow[lane + 32 * j], v);
        }
    }
}

// ---------------------------------------------------------------------------
// 3) score: per-node L1 norm of (feat - dst_norm * agg) over 96 features.
//    One wave per 16 nodes; the 16x96 |residual| tile is reduced against an
//    all-ones vector with 24 accumulating V_WMMA_F32_16X16X4_F32 ops.
//    A layout (32-bit 16x4): lanes 0-15 hold {K=4c+0, 4c+1} of row M=lane,
//    lanes 16-31 hold {K=4c+2, 4c+3} of row M=lane-16.
//    D layout (32-bit 16x16): VGPR v holds M=v (lanes 0-15) / M=v+8 (16-31);
//    all columns equal the row sum, so lane 0 emits rows 0-7, lane 16 rows 8-15.
// ---------------------------------------------------------------------------
__global__ void score_wmma_kernel(const float* __restrict__ feat,
                                  const float* __restrict__ agg,
                                  const float* __restrict__ in_deg,
                                  float* __restrict__ score,
                                  int n_nodes) {
    int lane = threadIdx.x & 31;
    int wave = (blockIdx.x * blockDim.x + threadIdx.x) >> 5;
    int base = wave * 16;
    if (base >= n_nodes) return;            // wave-uniform exit: EXEC stays all-1s

    int m    = lane & 15;
    int node = base + m;
    if (node >= n_nodes) node = n_nodes - 1;   // clamp (n_nodes % 16 == 0 here)
    int koff = (lane >> 4) * 2;             // 0 for lanes 0-15, 2 for lanes 16-31

    float norm = rsqrtf(fmaxf(in_deg[node], 1.0f));

    const float* __restrict__ frow = feat + (long long)node * D_FEAT;
    const float* __restrict__ arow = agg  + (long long)node * D_FEAT;

    v8f acc = {};
    v2f ones; ones.x = 1.0f; ones.y = 1.0f;

#pragma unroll
    for (int c = 0; c < D_FEAT / 4; ++c) {
        int k = 4 * c + koff;
        v2f a;
        a.x = fabsf(frow[k]     - arow[k]     * norm);
        a.y = fabsf(frow[k + 1] - arow[k + 1] * norm);
        // D = A(16x4) * ones(4x16) + C  ->  every column of D = row sum
        acc = __builtin_amdgcn_wmma_f32_16x16x4_f32(
            /*neg_a=*/false, a, /*neg_b=*/false, ones,
            /*c_mod=*/(short)0, acc, /*reuse_a=*/false, /*reuse_b=*/false);
    }

    if ((lane & 15) == 0) {
        int half = lane >> 4;               // 0 -> rows 0-7, 1 -> rows 8-15
#pragma unroll
        for (int v = 0; v < 8; ++v) {
            int row = base + half * 8 + v;
            if (row < n_nodes) score[row] = acc[v];
        }
    }
}

// ---------------------------------------------------------------------------
extern "C" void kernel_launch(void* const* d_in, const int* in_sizes, int n_in,
                              void* d_out, int out_size, void* d_ws, size_t ws_size,
                              hipStream_t stream) {
    const float* feat = (const float*)d_in[0];
    const int*   src  = (const int*)d_in[1];
    const int*   dst  = (const int*)d_in[2];
    float*       out  = (float*)d_out;

    int n_nodes = in_sizes[0] / D_FEAT;
    int n_edges = in_sizes[1];

    float* out_deg = (float*)d_ws;
    float* in_deg  = out_deg + n_nodes;
    float* agg     = in_deg  + n_nodes;

    // 0) zero degrees + agg
    long long nz = (long long)n_nodes * (2 + D_FEAT);
    {
        int threads = 256;
        int blocks  = (int)((nz + threads - 1) / threads);
        zero_kernel<<<blocks, threads, 0, stream>>>((float*)d_ws, nz);
    }

    // 1) degrees
    {
        int threads = 256;
        int blocks  = (n_edges + threads - 1) / threads;
        if (blocks > 4096) blocks = 4096;
        degree_kernel<<<blocks, threads, 0, stream>>>(src, dst, out_deg, in_deg, n_edges);
    }

    // 2) scatter SpMM (wave per edge)
    {
        int threads = 256;
        int blocks  = 4096;                 // 32768 waves, grid-stride over edges
        scatter_kernel<<<blocks, threads, 0, stream>>>(feat, src, dst, out_deg, agg, n_edges);
    }

    // 3) WMMA row-sum score (wave per 16 nodes)
    {
        int waves   = (n_nodes + 15) / 16;
        int threads = 256;
        int blocks  = (waves * 32 + threads - 1) / threads;
        score_wmma_kernel<<<blocks, threads, 0, stream>>>(feat, agg, in_deg, out, n_nodes);
    }
}